// IOCClassifier_18030272708868
// MI455X (gfx1250) — compile-verified
//
#include <hip/hip_runtime.h>

typedef __attribute__((ext_vector_type(2))) float v2f;
typedef __attribute__((ext_vector_type(8))) float v8f;

#define N_EVENT 10000
#define N_IOC   40000
#define NNODES  50000
#define NPAD    50176       // padded to 784*64; pad rows are inert (row-independent GEMM)
#define NE      800000
#define HD      256
#define NR      8
#define NB      8
#define NL      3
#define EDIM    16
#define LN_EPS  1e-5f

#define MROWS   64          // rows per layer-GEMM block (50176 = 784*64)
#define MTILES  4           // 16-row tiles per wave
#define LPITCH  260         // LDS row stride (floats): 16B-aligned rows, banks 4 apart

// ---------------------------------------------------------------- utilities

__global__ void zero_kernel(float4* p, long n4) {
    long i = (long)blockIdx.x * blockDim.x + threadIdx.x;
    if (i < n4) p[i] = make_float4(0.f, 0.f, 0.f, 0.f);
}

__global__ void copy_out_kernel(const float* __restrict__ h, float* __restrict__ out) {
    long i = (long)blockIdx.x * blockDim.x + threadIdx.x;
    if (i < (long)N_EVENT * HD) out[i] = h[i];
}

// W_r = sum_b comp[r,b] * bases[b]; slot 8 = root
__global__ void compute_w_kernel(const float* __restrict__ bases,
                                 const float* __restrict__ comp,
                                 const float* __restrict__ root,
                                 float* __restrict__ Wbuf) {
    int idx = blockIdx.x * blockDim.x + threadIdx.x;   // 9*65536
    int s = idx >> 16;
    int io = idx & 65535;
    if (s < NR) {
        float acc = 0.f;
        #pragma unroll
        for (int b = 0; b < NB; ++b)
            acc += comp[s * NB + b] * bases[(long)b * HD * HD + io];
        Wbuf[idx] = acc;
    } else {
        Wbuf[idx] = root[io];
    }
}

// --------------------------------------------------- edge MLP + aggregation

__global__ void edge_feat_kernel(const int* __restrict__ edge_index,
                                 const int* __restrict__ edge_type,
                                 const float* __restrict__ edge_weight,
                                 const float* __restrict__ emb,
                                 const float* __restrict__ mlpW,
                                 const float* __restrict__ mlpB,
                                 float* enh_sum, float* deg, float* cnt) {
    __shared__ float sW[17 * HD];
    __shared__ float sB[HD];
    __shared__ float sE[NR * EDIM];
    for (int i = threadIdx.x; i < 17 * HD; i += blockDim.x) sW[i] = mlpW[i];
    for (int i = threadIdx.x; i < HD; i += blockDim.x)      sB[i] = mlpB[i];
    for (int i = threadIdx.x; i < NR * EDIM; i += blockDim.x) sE[i] = emb[i];
    __syncthreads();

    int wave = threadIdx.x >> 5, lane = threadIdx.x & 31;
    int e = blockIdx.x * (blockDim.x >> 5) + wave;
    if (e >= NE) return;
    int ty = edge_type[e];
    int dst = edge_index[NE + e];
    float w = edge_weight[e];
    const float* er = sE + ty * EDIM;
    #pragma unroll
    for (int j = 0; j < 8; ++j) {
        int c = lane + 32 * j;
        float acc = sB[c] + w * sW[16 * HD + c];
        #pragma unroll
        for (int i = 0; i < EDIM; ++i) acc += er[i] * sW[i * HD + c];
        acc = fmaxf(acc, 0.f);
        atomicAdd(&enh_sum[(long)dst * HD + c], acc);
    }
    if (lane == 0) {
        atomicAdd(&deg[dst], 1.0f);
        atomicAdd(&cnt[dst * NR + ty], 1.0f);
    }
}

// enh := 0.1 * enh / max(deg,1)
__global__ void finalize_mean_kernel(float* enh, const float* __restrict__ deg) {
    long i = (long)blockIdx.x * blockDim.x + threadIdx.x;
    if (i < (long)NPAD * HD) {
        float d = fmaxf(deg[i >> 8], 1.0f);
        enh[i] = 0.1f * enh[i] / d;
    }
}

__global__ void inv_cnt_kernel(float* cnt) {
    int i = blockIdx.x * blockDim.x + threadIdx.x;
    if (i < NPAD * NR) cnt[i] = 1.0f / fmaxf(cnt[i], 1.0f);
}

// per-edge: t[dst, type, :] += h[src, :]
__global__ void scatter_kernel(const int* __restrict__ edge_index,
                               const int* __restrict__ edge_type,
                               const float* __restrict__ h,
                               float* t) {
    int wave = threadIdx.x >> 5, lane = threadIdx.x & 31;
    int e = blockIdx.x * (blockDim.x >> 5) + wave;
    if (e >= NE) return;
    int src = edge_index[e];
    int dst = edge_index[NE + e];
    int ty = edge_type[e];
    const float* hs = h + (long)src * HD;
    float* td = t + ((long)dst * NR + ty) * HD;
    #pragma unroll
    for (int j = 0; j < 8; ++j) {
        int c = lane + 32 * j;
        atomicAdd(&td[c], hs[c]);
    }
}

// ---------------------------------------------- async global->LDS tile stage
// 512 threads copy MROWS x 256 floats (64KB) as 8 b128 async loads per thread.
__device__ __forceinline__ void issue_group(const float* __restrict__ base,
                                            long gstride, float* buf, int tid) {
    #pragma unroll
    for (int i = 0; i < 8; ++i) {
        int idx = tid + 512 * i;
        int r = idx >> 6, q = idx & 63;
        const float* g = base + (long)r * gstride + (q << 2);
        unsigned lds = (unsigned)(size_t)&buf[r * LPITCH + (q << 2)];
        asm volatile("global_load_async_to_lds_b128 %0, %1, off"
                     :: "v"(lds), "v"(g) : "memory");
    }
}

// ------------------------------------------------ projection GEMM + LN + ReLU
template <int K>
__global__ void proj_gemm_kernel(const float* __restrict__ A,
                                 const float* __restrict__ W,
                                 const float* __restrict__ bias,
                                 const float* __restrict__ gamma,
                                 const float* __restrict__ beta,
                                 float* __restrict__ hOut, int rowOffset) {
    __shared__ float sA[16 * LPITCH];
    __shared__ float sOut[16 * HD];
    int tid = threadIdx.x;
    int lane = tid & 31, wave = tid >> 5;
    int half = lane >> 4, l16 = lane & 15;
    int n0 = blockIdx.x * 16;
    int colBase = wave * 16;
    const float* Wc = W + colBase + l16;

    constexpr int CW = (K < 256) ? K : 256;
    v8f c = {0.f, 0.f, 0.f, 0.f, 0.f, 0.f, 0.f, 0.f};
    for (int kb = 0; kb < K; kb += CW) {
        __syncthreads();
        #pragma unroll
        for (int idx = tid; idx < 16 * (CW / 2); idx += 512) {
            int r = idx / (CW / 2), kk = idx - r * (CW / 2);
            float2 v = *(const float2*)(A + (long)(n0 + r) * K + kb + kk * 2);
            *(float2*)&sA[r * LPITCH + kk * 2] = v;
        }
        __syncthreads();
        #pragma unroll 8
        for (int k = 0; k < CW; k += 4) {
            int ka = k + 2 * half;
            v2f a, b;
            a[0] = sA[l16 * LPITCH + ka];
            a[1] = sA[l16 * LPITCH + ka + 1];
            b[0] = Wc[(long)(kb + ka) * HD];
            b[1] = Wc[(long)(kb + ka + 1) * HD];
            c = __builtin_amdgcn_wmma_f32_16x16x4_f32(false, a, false, b,
                                                      (short)0, c, false, false);
        }
    }
    #pragma unroll
    for (int v = 0; v < 8; ++v)
        sOut[(v + 8 * half) * HD + colBase + l16] = c[v];
    __syncthreads();

    int row = wave;
    float x[8];
    float s1 = 0.f, s2 = 0.f;
    #pragma unroll
    for (int j = 0; j < 8; ++j) {
        int cc = lane + 32 * j;
        float xv = sOut[row * HD + cc] + bias[cc];
        x[j] = xv; s1 += xv; s2 += xv * xv;
    }
    #pragma unroll
    for (int off = 16; off >= 1; off >>= 1) {
        s1 += __shfl_xor(s1, off, 32);
        s2 += __shfl_xor(s2, off, 32);
    }
    float mu = s1 * (1.0f / HD);
    float var = s2 * (1.0f / HD) - mu * mu;
    float rstd = rsqrtf(var + LN_EPS);
    float* out = hOut + (long)(rowOffset + n0 + row) * HD;
    #pragma unroll
    for (int j = 0; j < 8; ++j) {
        int cc = lane + 32 * j;
        float y = (x[j] - mu) * rstd * gamma[cc] + beta[cc];
        out[cc] = fmaxf(y, 0.f);
    }
}

// -------------------------- RGCN layer: 9 accumulated GEMMs + fused epilogue
// Double-buffered async-to-LDS staging; per-source fragments scaled by 1/cnt
// after the K-loop (linear in A, so equivalent to scaling A rows).
__global__ void layer_gemm_kernel(const float* __restrict__ t,
                                  const float* __restrict__ invc,
                                  const float* __restrict__ h,
                                  const float* __restrict__ Wbuf,
                                  const float* __restrict__ bias,
                                  const float* __restrict__ enh,
                                  const float* __restrict__ gamma,
                                  const float* __restrict__ beta,
                                  float* __restrict__ hOut) {
    __shared__ __align__(16) float sBuf0[MROWS * LPITCH];
    __shared__ __align__(16) float sBuf1[MROWS * LPITCH];
    __shared__ float sInv[MROWS * NR];
    int tid = threadIdx.x;
    int lane = tid & 31, wave = tid >> 5;
    int half = lane >> 4, l16 = lane & 15;
    int n0 = blockIdx.x * MROWS;
    int colBase = wave * 16;

    for (int i = tid; i < MROWS * NR; i += 512) sInv[i] = invc[(long)n0 * NR + i];

    v8f ct[MTILES];
    #pragma unroll
    for (int m = 0; m < MTILES; ++m)
        ct[m] = (v8f){0.f, 0.f, 0.f, 0.f, 0.f, 0.f, 0.f, 0.f};

    issue_group(t + (long)n0 * NR * HD, (long)NR * HD, sBuf0, tid);   // source 0

    for (int s = 0; s < 9; ++s) {
        float* cur = (s & 1) ? sBuf1 : sBuf0;
        if (s < 8) {
            float* nxt = (s & 1) ? sBuf0 : sBuf1;
            if (s + 1 < 8)
                issue_group(t + ((long)n0 * NR + (s + 1)) * HD, (long)NR * HD, nxt, tid);
            else
                issue_group(h + (long)n0 * HD, (long)HD, nxt, tid);
            asm volatile("s_wait_asynccnt 0x8" ::: "memory");  // group s done (in-order)
        } else {
            asm volatile("s_wait_asynccnt 0x0" ::: "memory");
        }
        __syncthreads();    // group s visible to all waves

        v8f cs[MTILES];
        #pragma unroll
        for (int m = 0; m < MTILES; ++m)
            cs[m] = (v8f){0.f, 0.f, 0.f, 0.f, 0.f, 0.f, 0.f, 0.f};

        const float* Wc = Wbuf + (long)s * HD * HD + colBase + l16;
        #pragma unroll 4
        for (int k = 0; k < HD; k += 4) {
            int ka = k + 2 * half;
            v2f b;
            b[0] = Wc[ka * HD];
            b[1] = Wc[(ka + 1) * HD];
            #pragma unroll
            for (int m = 0; m < MTILES; ++m) {
                v2f a;
                int off = (m * 16 + l16) * LPITCH + ka;
                a[0] = cur[off];
                a[1] = cur[off + 1];
                cs[m] = __builtin_amdgcn_wmma_f32_16x16x4_f32(false, a, false, b,
                                                              (short)0, cs[m], false, false);
            }
        }
        if (s < NR) {
            #pragma unroll
            for (int m = 0; m < MTILES; ++m)
                #pragma unroll
                for (int v = 0; v < 8; ++v) {
                    int row = m * 16 + v + 8 * half;
                    ct[m][v] += sInv[row * NR + s] * cs[m][v];
                }
        } else {
            #pragma unroll
            for (int m = 0; m < MTILES; ++m)
                #pragma unroll
                for (int v = 0; v < 8; ++v)
                    ct[m][v] += cs[m][v];
        }
        __syncthreads();    // all waves done reading cur before it is re-filled
    }

    #pragma unroll
    for (int m = 0; m < MTILES; ++m)
        #pragma unroll
        for (int v = 0; v < 8; ++v)
            sBuf0[(m * 16 + v + 8 * half) * LPITCH + colBase + l16] = ct[m][v];
    __syncthreads();

    // LayerNorm + ReLU + residual: wave w handles rows w, w+16, w+32, w+48
    #pragma unroll
    for (int i = 0; i < MTILES; ++i) {
        int row = wave + 16 * i;
        long gRow = (long)(n0 + row) * HD;
        float x[8];
        float s1 = 0.f, s2 = 0.f;
        #pragma unroll
        for (int j = 0; j < 8; ++j) {
            int cc = lane + 32 * j;
            float xv = sBuf0[row * LPITCH + cc] + bias[cc] + enh[gRow + cc];
            x[j] = xv; s1 += xv; s2 += xv * xv;
        }
        #pragma unroll
        for (int off = 16; off >= 1; off >>= 1) {
            s1 += __shfl_xor(s1, off, 32);
            s2 += __shfl_xor(s2, off, 32);
        }
        float mu = s1 * (1.0f / HD);
        float var = s2 * (1.0f / HD) - mu * mu;
        float rstd = rsqrtf(var + LN_EPS);
        #pragma unroll
        for (int j = 0; j < 8; ++j) {
            int cc = lane + 32 * j;
            float y = (x[j] - mu) * rstd * gamma[cc] + beta[cc];
            hOut[gRow + cc] = fmaxf(y, 0.f) + h[gRow + cc];
        }
    }
}

// ----------------------------------------------------------------- launcher

extern "C" void kernel_launch(void* const* d_in, const int* in_sizes, int n_in,
                              void* d_out, int out_size, void* d_ws, size_t ws_size,
                              hipStream_t stream) {
    const float* x_event = (const float*)d_in[0];
    const float* x_ioc   = (const float*)d_in[1];
    const int*   edge_index = (const int*)d_in[2];
    const int*   edge_type  = (const int*)d_in[3];
    const float* edge_weight = (const float*)d_in[4];
    const float* pe_w = (const float*)d_in[5];
    const float* pe_b = (const float*)d_in[6];
    const float* pe_g = (const float*)d_in[7];
    const float* pe_beta = (const float*)d_in[8];
    const float* pi_w = (const float*)d_in[9];
    const float* pi_b = (const float*)d_in[10];
    const float* pi_g = (const float*)d_in[11];
    const float* pi_beta = (const float*)d_in[12];
    const float* emb = (const float*)d_in[13];
    const float* mlp_w = (const float*)d_in[14];
    const float* mlp_b = (const float*)d_in[15];
    const float* bases = (const float*)d_in[16];
    const float* comp = (const float*)d_in[17];
    const float* root = (const float*)d_in[18];
    const float* cbias = (const float*)d_in[19];
    const float* ngamma = (const float*)d_in[20];
    const float* nbeta = (const float*)d_in[21];

    float* ws = (float*)d_ws;
    float* h0 = ws;
    float* h1 = h0 + (long)NPAD * HD;
    float* enh = h1 + (long)NPAD * HD;
    float* deg = enh + (long)NPAD * HD;
    float* cnt = deg + NPAD;
    float* Wbuf = cnt + (long)NPAD * NR;
    float* tbuf = Wbuf + 9L * HD * HD;

    // zero h0 + h1 + enh + deg + cnt (contiguous; also clears inert pad rows)
    long z1 = 3L * NPAD * HD + NPAD + (long)NPAD * NR;
    zero_kernel<<<(unsigned)((z1 / 4 + 255) / 256), 256, 0, stream>>>((float4*)h0, z1 / 4);

    // input projections -> h0
    proj_gemm_kernel<768><<<N_EVENT / 16, 512, 0, stream>>>(x_event, pe_w, pe_b, pe_g, pe_beta, h0, 0);
    proj_gemm_kernel<128><<<N_IOC / 16, 512, 0, stream>>>(x_ioc, pi_w, pi_b, pi_g, pi_beta, h0, N_EVENT);

    // edge MLP aggregation + degree/count
    edge_feat_kernel<<<NE / 8, 256, 0, stream>>>(edge_index, edge_type, edge_weight,
                                                 emb, mlp_w, mlp_b, enh, deg, cnt);
    finalize_mean_kernel<<<(NPAD * HD) / 256, 256, 0, stream>>>(enh, deg);
    inv_cnt_kernel<<<(NPAD * NR) / 256, 256, 0, stream>>>(cnt);

    float* hc = h0;
    float* hn = h1;
    for (int l = 0; l < NL; ++l) {
        compute_w_kernel<<<(9 * HD * HD) / 256, 256, 0, stream>>>(
            bases + (long)l * NB * HD * HD, comp + l * NR * NB,
            root + (long)l * HD * HD, Wbuf);
        long tn4 = (long)NPAD * NR * HD / 4;
        zero_kernel<<<(unsigned)((tn4 + 255) / 256), 256, 0, stream>>>((float4*)tbuf, tn4);
        scatter_kernel<<<NE / 8, 256, 0, stream>>>(edge_index, edge_type, hc, tbuf);
        layer_gemm_kernel<<<NPAD / MROWS, 512, 0, stream>>>(
            tbuf, cnt, hc, Wbuf, cbias + l * HD, enh,
            ngamma + l * HD, nbeta + l * HD, hn);
        float* tmp = hc; hc = hn; hn = tmp;
    }

    copy_out_kernel<<<(N_EVENT * HD) / 256, 256, 0, stream>>>(hc, (float*)d_out);
}